// OTTProxy_11682311045463
// MI455X (gfx1250) — compile-verified
//
#include <hip/hip_runtime.h>
#include <hip/hip_bf16.h>
#include <math.h>

typedef float v2f __attribute__((ext_vector_type(2)));
typedef float v8f __attribute__((ext_vector_type(8)));

#define IN_GAIN_F 1.8197008586099834f /* 10^(5.2/20) */
#define LDST 180                      /* LDS row stride (floats), 176 cols + pad */

__device__ __forceinline__ float gelu_exact(float z) {
  return 0.5f * z * (1.0f + erff(z * 0.70710678118654752f));
}

__device__ __forceinline__ float kneef(float x, float thr, float slope, float k) {
  float d = x - thr;
  float h = d + 0.5f * k;
  float g = slope * h * h / (2.0f * k);
  return (fabsf(d) <= 0.5f * k) ? g : ((d > 0.5f * k) ? slope * d : 0.0f);
}

// ---------------- weight fragment pre-swizzle ----------------
// A-fragment (16x4 f32, ISA 7.12.2): lane<16 -> {A[M=lane][K0],A[M=lane][K0+1]},
// lane>=16 -> {A[M=lane-16][K0+2],A[M=lane-16][K0+3]}.
// wfrag index: ((((L*3+tap)*16 + c)*4 + m)*32 + lane)*2 + j
__global__ void prep_kernel(const float* __restrict__ w1, const float* __restrict__ w2,
                            const float* __restrict__ w3, const float* __restrict__ w4,
                            float* __restrict__ wfrag, float* __restrict__ wfrag4) {
  int idx = blockIdx.x * 256 + threadIdx.x;
  if (idx < 36864) {
    int L = idx / 12288; int r = idx % 12288;
    int tap = r / 4096;  r %= 4096;
    int c = r / 256;     r %= 256;
    int m = r / 64;      r %= 64;
    int lane = r >> 1;   int j = r & 1;
    int row = m * 16 + (lane & 15);
    int col = c * 4 + ((lane >> 4) << 1) + j;
    const float* w = (L == 0) ? w1 : ((L == 1) ? w2 : w3);
    wfrag[idx] = w[(row * 64 + col) * 3 + tap];
  } else if (idx < 36864 + 3072) {
    int r = idx - 36864;
    int tap = r / 1024;  r %= 1024;
    int c = r / 64;      r %= 64;
    int lane = r >> 1;   int j = r & 1;
    int row = lane & 15;
    int col = c * 4 + ((lane >> 4) << 1) + j;
    wfrag4[idx - 36864] = (row < 2) ? w4[(row * 64 + col) * 3 + tap] : 0.0f;
  }
}

// ---------------- dual FIR (512 taps, causal, zero left pad) ----------------
__global__ __launch_bounds__(256) void fir2_kernel(
    const float* __restrict__ x, const float* __restrict__ wl, const float* __restrict__ wh,
    float* __restrict__ yl, float* __restrict__ yh, int T, float gain) {
  __shared__ float xs[256 + 511];
  __shared__ float wls[512];
  __shared__ float whs[512];
  const int tid = threadIdx.x;
  const int bc = blockIdx.y;
  const int base = blockIdx.x * 256;
  for (int i = tid; i < 767; i += 256) {
    int t = base - 511 + i;
    xs[i] = (t >= 0 && t < T) ? x[(size_t)bc * T + t] * gain : 0.0f;
  }
  for (int i = tid; i < 512; i += 256) { wls[i] = wl[i]; whs[i] = wh[i]; }
  __syncthreads();
  int t = base + tid;
  if (t < T) {
    float al = 0.0f, ah = 0.0f;
#pragma unroll 8
    for (int j = 0; j < 512; ++j) {
      float xv = xs[tid + j];
      al = fmaf(wls[j], xv, al);
      ah = fmaf(whs[j], xv, ah);
    }
    yl[(size_t)bc * T + t] = al;
    yh[(size_t)bc * T + t] = ah;
  }
}

// ---------------- per-block RMS ----------------
__global__ void rms_kernel(const float* __restrict__ high, const float* __restrict__ mid,
                           const float* __restrict__ low, float* __restrict__ xrms,
                           int T, int Tc, int total) {
  int idx = blockIdx.x * blockDim.x + threadIdx.x;
  if (idx >= total) return;
  int i = idx / Tc, tc = idx % Tc;
  int b = i / 3, band = i % 3;            // band0=high, band1=mid, band2=low
  const float* p = (band == 0) ? high : ((band == 1) ? mid : low);
  const float* p0 = p + ((size_t)(b * 2 + 0)) * T + (size_t)tc * 64;
  const float* p1 = p + ((size_t)(b * 2 + 1)) * T + (size_t)tc * 64;
  float s = 0.0f;
  for (int j = 0; j < 64; ++j) {
    float a = p0[j], c = p1[j];
    s += 0.5f * (a * a + c * c);
  }
  xrms[idx] = sqrtf(s * (1.0f / 64.0f) + 1e-7f);
}

// ---------------- decaying cummax scan (env[n]=max(k*env[n-1], rms[n])) ----------------
__global__ void scan_kernel(const float* __restrict__ xrms, const float* __restrict__ rel,
                            float* __restrict__ envc, int Tc) {
  int i = threadIdx.x;
  if (i >= 24) return;
  float k = 1.0f - rel[i >> 3];  // replicates reference's repeat(rel,B) index quirk
  float e = 0.0f;
  const float* xr = xrms + (size_t)i * Tc;
  float* ec = envc + (size_t)i * Tc;
  for (int tc = 0; tc < Tc; ++tc) { e = fmaxf(e * k, xr[tc]); ec[tc] = e; }
}

// ---------------- gains + band recombine ----------------
__global__ void combine_kernel(const float* __restrict__ high, const float* __restrict__ mid,
                               const float* __restrict__ low, const float* __restrict__ envc,
                               const float* __restrict__ params, const float* __restrict__ kneep,
                               float* __restrict__ combined, int T, int Tc, int total) {
  int idx = blockIdx.x * blockDim.x + threadIdx.x;
  if (idx >= total) return;
  int b = idx / T, t = idx % T;
  float kdb = kneep[0];
  float pos = (float)t * ((float)(Tc - 1) / (float)(T - 1));
  int i0 = (int)floorf(pos);
  i0 = min(max(i0, 0), Tc - 2);
  float frac = pos - (float)i0;
  const float aslope_lo = 1.0f - 1.0f / 66.7f;
  const float bslope = 1.0f - 1.0f / 4.17f;
  float g[3];
#pragma unroll
  for (int band = 0; band < 3; ++band) {
    float ta = params[b * 7 + (3 - band)];
    float tb = params[b * 7 + (6 - band)];
    float above = (band == 0) ? 1.0f : aslope_lo;
    float og = (band == 1) ? 5.7f : 10.3f;
    const float* ec = envc + ((size_t)(b * 3 + band)) * Tc;
    float env = ec[i0] * (1.0f - frac) + ec[i0 + 1] * frac;
    float edb = 20.0f * log10f(env + 1e-7f);
    float gdb = -kneef(edb, ta, above, kdb) + kneef(-edb, -tb, bslope, kdb);
    float gg = fminf(fmaxf(gdb + og, -80.0f), 40.0f);
    g[band] = exp10f(gg * 0.05f);
  }
#pragma unroll
  for (int c = 0; c < 2; ++c) {
    size_t o = ((size_t)(b * 2 + c)) * T + t;
    combined[o] = high[o] * g[0] + mid[o] * g[1] + low[o] * g[2];
  }
}

// ---------------- fused dilated TCN via V_WMMA_F32_16X16X4_F32 ----------------
// One block per (batch, 128-sample tile). LDS buffer holds one layer's activations
// over columns cc in [0,176): global t = t0 - 24 + cc. Guards cc<8, cc>=168 are zero.
__global__ __launch_bounds__(256) void tcn_kernel(
    const float* __restrict__ audio, const float* __restrict__ params,
    const float* __restrict__ combined,
    const float* __restrict__ w0, const float* __restrict__ b0,
    const float* __restrict__ b1, const float* __restrict__ b2, const float* __restrict__ b3,
    const float* __restrict__ b4,
    const float* __restrict__ wfrag, const float* __restrict__ wfrag4,
    float* __restrict__ out, int T) {
  __shared__ float buf[64 * LDST];
  __shared__ float comb[2 * LDST];
  const int tid = threadIdx.x;
  const int wave = tid >> 5;
  const int lane = tid & 31;
  const int l15 = lane & 15;
  const int hi = lane >> 4;  // 0 or 1
  const int b = blockIdx.y;
  const int t0 = blockIdx.x * 128;

  // stage combined tile (zero outside [0,T)) + zero LDS guard columns
  for (int i = tid; i < 2 * 176; i += 256) {
    int ch = i / 176, cc = i % 176;
    int t = t0 - 24 + cc;
    comb[ch * LDST + cc] = (t >= 0 && t < T) ? combined[((size_t)(b * 2 + ch)) * T + t] : 0.0f;
  }
  for (int i = tid; i < 64 * 16; i += 256) {
    int row = i >> 4, g = i & 15;
    int cc = (g < 8) ? g : (160 + g);
    buf[row * LDST + cc] = 0.0f;
  }
  __syncthreads();

  // layer 0: 2 -> 64, dil 1 (VALU; K=2 too small for WMMA)
  for (int i = tid; i < 64 * 160; i += 256) {
    int oc = i / 160;
    int cc = 8 + (i % 160);
    int t = t0 - 24 + cc;
    float z = b0[oc];
#pragma unroll
    for (int ic = 0; ic < 2; ++ic)
#pragma unroll
      for (int k = 0; k < 3; ++k)
        z = fmaf(w0[(oc * 2 + ic) * 3 + k], comb[ic * LDST + cc - 1 + k], z);
    buf[oc * LDST + cc] = (t >= 0 && t < T) ? gelu_exact(z) : 0.0f;
  }
  __syncthreads();

  // layers 1..3: 64 -> 64, dil 2/4/8, WMMA fp32 16x16x4, in-place via reg staging
  for (int L = 0; L < 3; ++L) {
    const int dil = 2 << L;
    const float* bias = (L == 0) ? b1 : ((L == 1) ? b2 : b3);
    v8f acc[5];
#pragma unroll
    for (int jj = 0; jj < 5; ++jj) {
      int ti = wave + 8 * jj;          // 40 tiles over 8 waves
      int m = ti & 3, n = ti >> 2;     // m: out-ch tile, n: time tile
      v8f a;
#pragma unroll
      for (int v = 0; v < 8; ++v) a[v] = bias[m * 16 + v + hi * 8];
      acc[jj] = a;
#pragma unroll
      for (int tap = 0; tap < 3; ++tap) {
        int col = 8 + n * 16 + (tap - 1) * dil + l15;   // stays in [0,176)
#pragma unroll
        for (int c = 0; c < 16; ++c) {
          v2f av = *(const v2f*)(wfrag + ((((L * 3 + tap) * 16 + c) * 4 + m) * 64) + lane * 2);
          int rk = c * 4 + hi * 2;
          v2f bv;
          bv.x = buf[rk * LDST + col];
          bv.y = buf[(rk + 1) * LDST + col];
          acc[jj] = __builtin_amdgcn_wmma_f32_16x16x4_f32(false, av, false, bv,
                                                          (short)0, acc[jj], false, false);
        }
      }
    }
    __syncthreads();   // all reads of layer L done before overwrite
#pragma unroll
    for (int jj = 0; jj < 5; ++jj) {
      int ti = wave + 8 * jj;
      int m = ti & 3, n = ti >> 2;
      int col = 8 + n * 16 + l15;
      int t = t0 - 24 + col;
      bool ok = (t >= 0 && t < T);
#pragma unroll
      for (int v = 0; v < 8; ++v) {
        int row = m * 16 + v + hi * 8;
        buf[row * LDST + col] = ok ? gelu_exact(acc[jj][v]) : 0.0f;
      }
    }
    __syncthreads();
  }

  // layer 4: 64 -> 2 (rows >= 2 padded zero), dil 1, then dry/wet mix
  {
    int n = wave;                       // 8 waves x 16 cols = 128 outputs
    int colbase0 = 24 + n * 16;
    v8f acc;
#pragma unroll
    for (int v = 0; v < 8; ++v) {
      int row = v + hi * 8;
      acc[v] = (row < 2) ? b4[row] : 0.0f;
    }
#pragma unroll
    for (int tap = 0; tap < 3; ++tap) {
      int col = colbase0 + (tap - 1) + l15;
#pragma unroll
      for (int c = 0; c < 16; ++c) {
        v2f av = *(const v2f*)(wfrag4 + ((tap * 16 + c) * 64) + lane * 2);
        int rk = c * 4 + hi * 2;
        v2f bv;
        bv.x = buf[rk * LDST + col];
        bv.y = buf[(rk + 1) * LDST + col];
        acc = __builtin_amdgcn_wmma_f32_16x16x4_f32(false, av, false, bv,
                                                    (short)0, acc, false, false);
      }
    }
    float amt = params[b * 7 + 0];
    int t = t0 + n * 16 + l15;
    if (hi == 0 && t < T) {             // lanes 0..15 hold rows 0..7 -> rows 0,1 valid
      int cc = colbase0 + l15;
#pragma unroll
      for (int ch = 0; ch < 2; ++ch) {
        size_t o = ((size_t)(b * 2 + ch)) * T + t;
        float val = (1.0f - amt) * audio[o] + amt * (comb[ch * LDST + cc] + acc[ch]);
        out[o] = val;
      }
    }
  }
}

extern "C" void kernel_launch(void* const* d_in, const int* in_sizes, int n_in,
                              void* d_out, int out_size, void* d_ws, size_t ws_size,
                              hipStream_t stream) {
  const float* audio = (const float*)d_in[0];
  const float* params = (const float*)d_in[1];
  const float* rel = (const float*)d_in[2];
  const float* kneep = (const float*)d_in[3];
  const float* ir_ll = (const float*)d_in[4];
  const float* ir_lh = (const float*)d_in[5];
  const float* ir_hl = (const float*)d_in[6];
  const float* ir_hh = (const float*)d_in[7];
  const float* w0 = (const float*)d_in[8];
  const float* b0 = (const float*)d_in[9];
  const float* w1 = (const float*)d_in[10];
  const float* b1 = (const float*)d_in[11];
  const float* w2 = (const float*)d_in[12];
  const float* b2 = (const float*)d_in[13];
  const float* w3 = (const float*)d_in[14];
  const float* b3 = (const float*)d_in[15];
  const float* w4 = (const float*)d_in[16];
  const float* b4 = (const float*)d_in[17];

  const int B = in_sizes[1] / 7;        // 8
  const int T = in_sizes[0] / (2 * B);  // 88200
  const int Tc = T / 64;                // 1378
  const size_t N = (size_t)B * 2 * T;

  float* ws = (float*)d_ws;
  float* m_h = ws;
  float* low = ws + N;
  float* mid = ws + 2 * N;
  float* high = ws + 3 * N;
  float* combined = ws + 4 * N;
  float* xrms = ws + 5 * N;
  float* envc = xrms + (size_t)B * 3 * Tc;
  float* wfrag = envc + (size_t)B * 3 * Tc;
  float* wfrag4 = wfrag + 36864;

  prep_kernel<<<(36864 + 3072 + 255) / 256, 256, 0, stream>>>(w1, w2, w3, w4, wfrag, wfrag4);

  dim3 gfir((T + 255) / 256, B * 2);
  fir2_kernel<<<gfir, 256, 0, stream>>>(audio, ir_ll, ir_lh, low, m_h, T, IN_GAIN_F);
  fir2_kernel<<<gfir, 256, 0, stream>>>(m_h, ir_hl, ir_hh, mid, high, T, 1.0f);

  int totrms = B * 3 * Tc;
  rms_kernel<<<(totrms + 255) / 256, 256, 0, stream>>>(high, mid, low, xrms, T, Tc, totrms);
  scan_kernel<<<1, 32, 0, stream>>>(xrms, rel, envc, Tc);

  int totc = B * T;
  combine_kernel<<<(totc + 255) / 256, 256, 0, stream>>>(high, mid, low, envc, params, kneep,
                                                         combined, T, Tc, totc);

  dim3 gt((T + 127) / 128, B);
  tcn_kernel<<<gt, 256, 0, stream>>>(audio, params, combined, w0, b0, b1, b2, b3, b4,
                                     wfrag, wfrag4, (float*)d_out, T);
}